// SpanEndpointsV2_90099823935817
// MI455X (gfx1250) — compile-verified
//
#include <hip/hip_runtime.h>

// Problem constants (from reference: x = (4, 2048, 768) f32, K = 12)
#define BK     12            // span width K
#define LL     2048          // sequence length
#define DD     768           // feature dim
#define TILE_L 16            // rows per block
#define HALO   (BK - 1)      // 11 extra rows for span ends
#define ROWS   (TILE_L + HALO)           // 27 rows staged in LDS
#define SMEM_BYTES (ROWS * DD * 4)       // 82,944 B  (< 320 KB/WGP)
#define THREADS 256                      // 8 wave32 waves

typedef float v4f __attribute__((ext_vector_type(4)));

__device__ __forceinline__ float logaddexp_f32(float a, float b) {
  // logaddexp(a,b) = max(a,b) + log1p(exp(-|a-b|)), in base-2 HW units
  float m = fmaxf(a, b);
  float d = -fabsf(a - b);
  float t = __builtin_amdgcn_exp2f(d * 1.44269504088896340736f);   // v_exp_f32
  return m + __builtin_amdgcn_logf(1.0f + t) * 0.69314718055994530942f; // v_log_f32
}

__global__ __launch_bounds__(THREADS)
void span_logaddexp_kernel(const float* __restrict__ x, float* __restrict__ out) {
  __shared__ float smem[ROWS * DD];

  const int tiles_per_b = LL / TILE_L;
  const int b  = blockIdx.x / tiles_per_b;
  const int l0 = (blockIdx.x % tiles_per_b) * TILE_L;

  // ---------------- Stage tile + halo into LDS with async global->LDS DMA ----
  const unsigned long long gbase =
      (unsigned long long)(x + ((size_t)b * LL + l0) * DD);
  const int valid_rows  = (LL - l0) < ROWS ? (LL - l0) : ROWS;
  const int valid_bytes = valid_rows * DD * 4;   // multiple of 16
  const unsigned lds0   = (unsigned)(uintptr_t)&smem[0]; // low 32b = LDS offset

  // Branch-free async-copy clause over the valid region (ASYNCcnt-tracked)
  for (int ofs = threadIdx.x * 16; ofs < valid_bytes; ofs += THREADS * 16) {
    unsigned laddr = lds0 + (unsigned)ofs;
    unsigned goff  = (unsigned)ofs;
    asm volatile("global_load_async_to_lds_b128 %0, %1, %2"
                 :: "v"(laddr), "v"(goff), "s"(gbase) : "memory");
  }
  // Zero-pad rows past end of sequence (non-empty only for last tile per batch)
  for (int ofs = valid_bytes + (int)threadIdx.x * 16; ofs < SMEM_BYTES;
       ofs += THREADS * 16) {
    v4f z = (v4f)0.0f;
    *(v4f*)&smem[ofs >> 2] = z;
  }
  asm volatile("s_wait_asynccnt 0x0" ::: "memory");
  __syncthreads();

  // ---------------- Compute 12 logaddexp spans per staged row ----------------
  float* outb = out + (((size_t)b * LL + l0) * BK) * DD;

  // Each thread owns (l, d4) pairs: consecutive lanes -> consecutive d4 within
  // one row, so every (l,k) row store is a coalesced 512B wave write.
  for (int p = threadIdx.x; p < TILE_L * (DD / 4); p += THREADS) {
    const int d4 = p % (DD / 4);
    const int l  = p / (DD / 4);
    const v4f s = *(const v4f*)&smem[l * DD + d4 * 4];
#pragma unroll
    for (int k = 0; k < BK; ++k) {
      const v4f e = *(const v4f*)&smem[(l + k) * DD + d4 * 4];
      v4f r;
      r.x = logaddexp_f32(s.x, e.x);
      r.y = logaddexp_f32(s.y, e.y);
      r.z = logaddexp_f32(s.z, e.z);
      r.w = logaddexp_f32(s.w, e.w);
      float* dst = outb + ((size_t)l * BK + k) * DD + d4 * 4;
      // 302 MB output stream > 192 MB L2: bypass with non-temporal store
      __builtin_nontemporal_store(r, (v4f*)dst);
    }
  }
}

// span_idx[l,k] = (l, l+k); d_out is one flat float32 buffer for the tuple,
// so indices are written as their (exactly representable) float values.
__global__ __launch_bounds__(256)
void span_idx_kernel(float* __restrict__ out2) {
  int i = blockIdx.x * blockDim.x + threadIdx.x;   // over L*K
  if (i < LL * BK) {
    int l = i / BK;
    int k = i % BK;
    out2[2 * i + 0] = (float)l;
    out2[2 * i + 1] = (float)(l + k);
  }
}

extern "C" void kernel_launch(void* const* d_in, const int* in_sizes, int n_in,
                              void* d_out, int out_size, void* d_ws, size_t ws_size,
                              hipStream_t stream) {
  (void)in_sizes; (void)n_in; (void)d_ws; (void)ws_size; (void)out_size;
  const float* x = (const float*)d_in[0];
  float* out = (float*)d_out;

  const int B = 4;
  dim3 grid(B * (LL / TILE_L));                    // 512 blocks
  span_logaddexp_kernel<<<grid, THREADS, 0, stream>>>(x, out);

  const size_t reps_elems = (size_t)B * LL * BK * DD;   // 75,497,472
  const int idx_items = LL * BK;                        // 24,576
  span_idx_kernel<<<(idx_items + 255) / 256, 256, 0, stream>>>(out + reps_elems);
}